// BaseHysteresis_26465588478417
// MI455X (gfx1250) — compile-verified
//
#include <hip/hip_runtime.h>
#include <math.h>

#define SEQ_LEN 2048
#define M_HYST  20100          // 200*201/2 hysterons
#define NW      640            // padded wave count (multiple of 4), 640*32 = 20480 threads
#define KSCALE  144.269504088896f   // (1/TEMP) * log2(e) = 100 * 1.4426950408...

typedef float v2f __attribute__((ext_vector_type(2)));
typedef float v8f __attribute__((ext_vector_type(8)));

__device__ __forceinline__ float softplus_f(float z) {
    // log1p(exp(z)), numerically stable like jax.nn.softplus
    return (z > 20.0f) ? z : log1pf(expf(z));
}

// --- Kernel 1: normalized field sequence norm_h[t] = (x[t]-h_min)/h_range ---
__global__ void hyst_prep_kernel(const float* __restrict__ x,
                                 const float* __restrict__ h_min,
                                 const float* __restrict__ h_range,
                                 float* __restrict__ ws_h) {
    int t = blockIdx.x * blockDim.x + threadIdx.x;
    if (t < SEQ_LEN) ws_h[t] = (x[t] - h_min[0]) / h_range[0];
}

// --- Kernel 2: sum of softplus(raw_density), single-block deterministic reduce ---
__global__ void hyst_sumdens_kernel(const float* __restrict__ raw_density,
                                    float* __restrict__ ws_s) {
    __shared__ float red[256];
    float s = 0.0f;
    for (int i = threadIdx.x; i < M_HYST; i += 256) s += softplus_f(raw_density[i]);
    red[threadIdx.x] = s;
    __syncthreads();
    for (int off = 128; off > 0; off >>= 1) {
        if (threadIdx.x < (unsigned)off) red[threadIdx.x] += red[threadIdx.x + off];
        __syncthreads();
    }
    if (threadIdx.x == 0) ws_s[0] = red[0];
}

// --- Kernel 3: the sequential Preisach scan. One hysteron per thread, loop over
// all 2048 steps; per step, wave-level shuffle reduction of state*dens partials,
// lane 0 of each wave writes its column of the [SEQ_LEN x NW] partial matrix. ---
__global__ __launch_bounds__(256) void hyst_scan_kernel(
    const float* __restrict__ mesh,        // [M,2] = (beta, alpha)
    const float* __restrict__ raw_density, // [M]
    const float* __restrict__ state0,      // [M]
    const float* __restrict__ field0,      // [1]
    const float* __restrict__ ws_h,        // [SEQ_LEN]
    float* __restrict__ ws_part)           // [SEQ_LEN * NW]
{
    const int gid   = blockIdx.x * blockDim.x + threadIdx.x;
    const int lane  = threadIdx.x & 31;
    const int gwave = gid >> 5;
    const bool active = (gid < M_HYST);

    float beta  = active ? mesh[2 * gid + 0] : 0.0f;
    float alpha = active ? mesh[2 * gid + 1] : 0.0f;
    float dens  = active ? softplus_f(raw_density[gid]) : 0.0f;
    float state = active ? state0[gid] : 0.0f;

    const float alpha_s = alpha * KSCALE;   // precompute (alpha/temp)*log2e
    const float beta_s  = beta  * KSCALE;
    float h_prev = field0[0];

    for (int t = 0; t < SEQ_LEN; ++t) {
        float h  = ws_h[t];                 // uniform scalar load
        float hs = h * KSCALE;
        // sigmoid((h-a)/temp) = rcp(1 + 2^((a-h)*100*log2e))
        float ea = __builtin_amdgcn_exp2f(alpha_s - hs);
        float eb = __builtin_amdgcn_exp2f(beta_s  - hs);
        float sa = __builtin_amdgcn_rcpf(1.0f + ea);
        float sb = __builtin_amdgcn_rcpf(1.0f + eb);
        float up = state + (1.0f - state) * sa;
        float dn = -1.0f + (state + 1.0f) * sb;
        float ns = (h > h_prev) ? up : ((h < h_prev) ? dn : state);
        state  = ns;
        h_prev = h;

        float p = state * dens;             // partial dot contribution
        p += __shfl_xor(p, 16, 32);
        p += __shfl_xor(p,  8, 32);
        p += __shfl_xor(p,  4, 32);
        p += __shfl_xor(p,  2, 32);
        p += __shfl_xor(p,  1, 32);
        if (lane == 0) ws_part[(size_t)t * NW + gwave] = p;
    }
}

// --- Kernel 4: WMMA row-sum of the [2048 x 640] partial matrix + fused epilogue.
// One wave per 16-timestep tile (128 waves). A = 16x4 f32 tile of partials
// (M = lane%16, K pair picked by lane/16 per the CDNA5 f32 A layout); B = all
// ones, so C[m,*] accumulates the row sum regardless of K-slot permutation. ---
__global__ __launch_bounds__(256) void hyst_reduce_wmma_kernel(
    const float* __restrict__ ws_part,
    const float* __restrict__ ws_h,
    const float* __restrict__ ws_s,
    const float* __restrict__ offset,
    const float* __restrict__ scale,
    const float* __restrict__ slope,
    float* __restrict__ out)
{
    const int lane = threadIdx.x & 31;
    const int wave = (blockIdx.x * blockDim.x + threadIdx.x) >> 5;
    const int t0   = wave * 16;                 // 128 waves * 16 rows = 2048
    const int m    = lane & 15;                 // A-matrix row M
    const int kb   = (lane >> 4) * 2;           // K base: lanes 0-15 -> K0/K1, 16-31 -> K2/K3
    const float* rowp = ws_part + (size_t)(t0 + m) * NW + kb;

    v8f c = {};
    v2f b; b.x = 1.0f; b.y = 1.0f;              // ones vector -> row sums
    for (int k0 = 0; k0 < NW; k0 += 4) {
        v2f a;
        a.x = rowp[k0];
        a.y = rowp[k0 + 1];
        c = __builtin_amdgcn_wmma_f32_16x16x4_f32(
                /*neg_a=*/false, a, /*neg_b=*/false, b,
                /*c_mod=*/(short)0, c, /*reuse_a=*/false, /*reuse_b=*/false);
    }

    const float invS = 1.0f / ws_s[0];
    const float sc = scale[0], of = offset[0], sl = slope[0];
    // C/D layout: VGPR r holds row M=r (lanes 0-15) or M=8+r (lanes 16-31);
    // all 16 columns identical, so lanes 0 and 16 write the 16 outputs.
    if ((lane & 15) == 0) {
        const int mb = (lane >> 4) * 8;
        for (int r = 0; r < 8; ++r) {
            int t = t0 + mb + r;
            out[t] = sc * (c[r] * invS) + of + sl * ws_h[t];
        }
    }
}

extern "C" void kernel_launch(void* const* d_in, const int* in_sizes, int n_in,
                              void* d_out, int out_size, void* d_ws, size_t ws_size,
                              hipStream_t stream) {
    const float* x       = (const float*)d_in[0];
    const float* mesh    = (const float*)d_in[1];
    const float* rawd    = (const float*)d_in[2];
    const float* offset  = (const float*)d_in[3];
    const float* scale   = (const float*)d_in[4];
    const float* slope   = (const float*)d_in[5];
    const float* state0  = (const float*)d_in[6];
    const float* field0  = (const float*)d_in[7];
    const float* h_min   = (const float*)d_in[8];
    const float* h_range = (const float*)d_in[9];

    float* wsf     = (float*)d_ws;
    float* ws_h    = wsf;            // 2048 floats
    float* ws_s    = wsf + 2048;     // 1 float (padded to 64)
    float* ws_part = wsf + 2112;     // 2048 * 640 floats (~5.24 MB, L2-resident)

    hyst_prep_kernel<<<8, 256, 0, stream>>>(x, h_min, h_range, ws_h);
    hyst_sumdens_kernel<<<1, 256, 0, stream>>>(rawd, ws_s);
    hyst_scan_kernel<<<NW / 8, 256, 0, stream>>>(mesh, rawd, state0, field0, ws_h, ws_part);
    hyst_reduce_wmma_kernel<<<16, 256, 0, stream>>>(ws_part, ws_h, ws_s,
                                                    offset, scale, slope, (float*)d_out);
}